// AvgPoolingLayer_81664508166880
// MI455X (gfx1250) — compile-verified
//
#include <hip/hip_runtime.h>

typedef __attribute__((ext_vector_type(2))) float v2f;
typedef __attribute__((ext_vector_type(8))) float v8f;

#define DCOLS 256
#define CHUNK 1024
#define NTHREADS 512   // 16 waves of 32
#define SENTINEL 0x3FFFFFFF

// ---------------------------------------------------------------------------
// Zero the output accumulator (harness poisons d_out; atomic accumulation
// needs a clean slate every call).
// ---------------------------------------------------------------------------
__global__ void zero_out_kernel(float* __restrict__ out, int total) {
    int idx = blockIdx.x * blockDim.x + threadIdx.x;
    if (idx < total) out[idx] = 0.0f;
}

// ---------------------------------------------------------------------------
// Per-segment row counts via binary search on the sorted int64 segment ids.
// counts[s] = lower_bound(s+1) - lower_bound(s)
// ---------------------------------------------------------------------------
__global__ void count_kernel(const long long* __restrict__ ids,
                             float* __restrict__ counts, int N, int S) {
    int s = blockIdx.x * blockDim.x + threadIdx.x;
    if (s >= S) return;
    long long key_lo = (long long)s;
    long long key_hi = (long long)s + 1;
    int lo0 = 0, hi0 = N;
    while (lo0 < hi0) { int mid = (lo0 + hi0) >> 1; if (ids[mid] < key_lo) lo0 = mid + 1; else hi0 = mid; }
    int lo1 = 0, hi1 = N;
    while (lo1 < hi1) { int mid = (lo1 + hi1) >> 1; if (ids[mid] < key_hi) lo1 = mid + 1; else hi1 = mid; }
    counts[s] = (float)(lo1 - lo0);
}

// ---------------------------------------------------------------------------
// Segment-sum via V_WMMA_F32_16X16X4_F32.
//   D = A x B + C, A = 16x4 one-hot segment selector, B = 4x16 feats tile,
//   C = 16 local segments x 16 cols fp32 accumulator (8 VGPRs).
// 16 waves per block, each owning a distinct 16-column tile of D=256;
// together they read each 1KB row fully coalesced. C flushes to global
// atomics only when the sorted-id 16-segment window advances (~1/1600 rows).
// All control decisions come from LDS-staged ids -> wave-uniform, so EXEC
// stays all-ones around the WMMAs as the ISA requires.
// ---------------------------------------------------------------------------
__global__ __launch_bounds__(NTHREADS)
void segsum_wmma_kernel(const float* __restrict__ feats,
                        const long long* __restrict__ ids,
                        float* __restrict__ out, int N, int S) {
    __shared__ alignas(16) int sid[CHUNK];

    const int tid  = threadIdx.x;
    const int lane = tid & 31;
    const int wave = tid >> 5;
    const int rowbase = blockIdx.x * CHUNK;

    // Stage this chunk's segment ids into LDS (int32 is plenty: S = 10k).
    for (int i = tid; i < CHUNK; i += NTHREADS) {
        int r = rowbase + i;
        sid[i] = (r < N) ? (int)ids[r] : SENTINEL;
    }
    __syncthreads();

    const int colbase = wave << 4;            // wave's 16-column tile
    const int n       = lane & 15;            // column within tile (B/C layout)
    const int m       = lane & 15;            // local segment row (A layout)
    const int khalf   = (lane >> 4) << 1;     // lanes 0-15 -> K=0,1; 16-31 -> K=2,3
    const int moff    = (lane >> 4) << 3;     // C layout: lanes 16-31 hold M=r+8

    v8f c = {};                               // 16x16 fp32 accumulator
    int base = sid[0];                        // current 16-segment window base

    auto flushC = [&]() {
#pragma unroll
        for (int r = 0; r < 8; ++r) {
            int s = base + r + moff;
            float v = c[r];
            if (s < S && v != 0.0f)
                atomicAdd(out + (unsigned)s * DCOLS + colbase + n, v);
        }
    };

    // One 16x16x4 WMMA covering rows [g, g+4). ia..id_ are those rows' ids.
    auto wmma4 = [&](int g, int ia, int ib, int ic, int id_, bool clampRows) {
        int idk0 = (lane < 16) ? ia : ic;
        int idk1 = (lane < 16) ? ib : id_;
        v2f a;                                 // one-hot selector
        a.x = (idk0 - base == m) ? 1.0f : 0.0f;
        a.y = (idk1 - base == m) ? 1.0f : 0.0f;
        int r0 = rowbase + g + khalf;
        int r1 = r0 + 1;
        if (clampRows) {                       // tail block only; A row is 0 there
            r0 = (r0 < N) ? r0 : (N - 1);
            r1 = (r1 < N) ? r1 : (N - 1);
        }
        v2f b;                                 // streamed exactly once -> NT hint
        b.x = __builtin_nontemporal_load(feats + (unsigned)r0 * DCOLS + colbase + n);
        b.y = __builtin_nontemporal_load(feats + (unsigned)r1 * DCOLS + colbase + n);
        // (neg_a, A, neg_b, B, c_mod, C, reuse_a, reuse_b)
        c = __builtin_amdgcn_wmma_f32_16x16x4_f32(
                false, a, false, b, (short)0, c, false, false);
    };

    auto group8 = [&](int g, bool clampRows) {
        const int4* sid4 = (const int4*)sid;   // 2x ds_load_b128, broadcast
        int4 q0 = sid4[(g >> 2)];
        int4 q1 = sid4[(g >> 2) + 1];
        if (q1.w > base + 15) {                // window must advance (ids sorted)
            flushC();
            c = (v8f){0.f, 0.f, 0.f, 0.f, 0.f, 0.f, 0.f, 0.f};
            base = q0.x;
            if (q1.w > base + 15) {
                // Pathological: 8 rows span >16 segment ids. Direct atomics.
#pragma unroll
                for (int k = 0; k < 8; ++k) {
                    int id = sid[g + k];
                    if (id < S && lane < 16) {
                        int r = rowbase + g + k;
                        atomicAdd(out + (unsigned)id * DCOLS + colbase + n,
                                  feats[(unsigned)r * DCOLS + colbase + n]);
                    }
                }
                base = q1.w;
                return;
            }
        }
        wmma4(g,     q0.x, q0.y, q0.z, q0.w, clampRows);
        wmma4(g + 4, q1.x, q1.y, q1.z, q1.w, clampRows);
    };

    const int nvalid = (N - rowbase < CHUNK) ? (N - rowbase) : CHUNK;
    const int gfull  = nvalid & ~7;            // leading groups with all rows valid
    for (int g = 0; g < gfull; g += 8)      group8(g, false);  // fast path
    for (int g = gfull; g < CHUNK; g += 8)  group8(g, true);   // tail block only
    flushC();
}

// ---------------------------------------------------------------------------
// out[s, d] = sum[s, d] / max(count[s], 1)
// ---------------------------------------------------------------------------
__global__ void divide_kernel(float* __restrict__ out,
                              const float* __restrict__ counts, int total) {
    int idx = blockIdx.x * blockDim.x + threadIdx.x;
    if (idx >= total) return;
    int s = idx >> 8;                          // / DCOLS
    out[idx] /= fmaxf(counts[s], 1.0f);
}

// ---------------------------------------------------------------------------
extern "C" void kernel_launch(void* const* d_in, const int* in_sizes, int n_in,
                              void* d_out, int out_size, void* d_ws, size_t ws_size,
                              hipStream_t stream) {
    const float*     feats = (const float*)d_in[0];
    const long long* ids   = (const long long*)d_in[1];
    float*           out   = (float*)d_out;
    float*           cnts  = (float*)d_ws;     // S floats of scratch

    const int N = in_sizes[1];                 // number of rows
    const int S = out_size / DCOLS;            // number of segments

    zero_out_kernel<<<(out_size + 1023) / 1024, 1024, 0, stream>>>(out, out_size);
    count_kernel<<<(S + 255) / 256, 256, 0, stream>>>(ids, cnts, N, S);

    const int nblocks = (N + CHUNK - 1) / CHUNK;
    segsum_wmma_kernel<<<nblocks, NTHREADS, 0, stream>>>(feats, ids, out, N, S);

    divide_kernel<<<(out_size + 255) / 256, 256, 0, stream>>>(out, cnts, out_size);
}